// SRMotionGate_49031346651729
// MI455X (gfx1250) — compile-verified
//
#include <hip/hip_runtime.h>
#include <hip/hip_bf16.h>
#include <math.h>

typedef __attribute__((ext_vector_type(2))) float v2f;
typedef __attribute__((ext_vector_type(8))) float v8f;

// D = A(16x4 f32) x B(4x16 f32) + C(16x16 f32), exact fp32 accumulate.
#define WMMA_F32(a, bv, c) \
  __builtin_amdgcn_wmma_f32_16x16x4_f32(false, (a), false, (bv), (short)0, (c), false, false)

// Problem constants (B,N,T,H,L)=(32,128,2,64,2), H1=32, H2=64
constexpr int Nn = 128, Tt = 2, Hh = 64, H1 = 32, H2 = 64;

__global__ __launch_bounds__(256) void srmg_kernel(
    const float* __restrict__ trace, const float* __restrict__ cell,
    const float* __restrict__ hid,
    const float* __restrict__ W1, const float* __restrict__ b1,
    const float* __restrict__ W2, const float* __restrict__ b2,
    const float* __restrict__ W3, const float* __restrict__ b3,
    const float* __restrict__ Wu, const float* __restrict__ Wg,
    const float* __restrict__ bg, const float* __restrict__ Wc,
    float* __restrict__ out)
{
  __shared__ float sY[Nn * H1];        // 16KB  y = trace @ W1^T  (per batch)
  __shared__ float sHu1[Nn];           // h . wu[64:128]   (p term)
  __shared__ float sHu2[Nn];           // h . wu[128:192]  (l term)
  __shared__ float sGbP[16 * Hh];      // 4KB   (h[p] @ Gb^T + bg) for p-tile
  __shared__ float sHl[16 * Hh];       // 4KB   h[l] tile (per l-tile)
  __shared__ float sGcL[16 * Hh];      // 4KB   h[l] @ Gc^T tile
  __shared__ float sU[16 * Nn];        // 8KB   u -> alpha
  __shared__ float sAgg[16 * Hh];      // 4KB   aggregated messages
  __shared__ float sW2[H2 * H1];       // 8KB
  __shared__ float sW3[Hh * H2];       // 16KB
  __shared__ float sGa[Hh * Hh];       // 16KB  Wg[:, :64]
  __shared__ float sB1[H1], sB2[H2], sB3[Hh], sWu0[Hh];
  __shared__ float sStage[8 * 16 * Hh]; // 32KB  per-wave GEMM re-layout stage

  const int tid  = threadIdx.x;
  const int wave = tid >> 5;
  const int lane = tid & 31;
  const int b    = blockIdx.x >> 3;       // batch
  const int p0   = (blockIdx.x & 7) * 16; // p-tile base

  const int m  = lane & 15;   // row within 16-tile (ll) / col selector
  const int hi = lane >> 4;   // lane half
  const int kb = hi * 2;      // K-pair base within 4-wide chunk
  float* stage = &sStage[wave * 16 * Hh];

  // ---------------- setup: weights + per-node precomputes into LDS ----------
  for (int i = tid; i < H2 * H1; i += 256) sW2[i] = W2[i];
  for (int i = tid; i < Hh * H2; i += 256) sW3[i] = W3[i];
  for (int i = tid; i < Hh * Hh; i += 256) sGa[i] = Wg[(i >> 6) * 192 + (i & 63)];
  if (tid < H1) sB1[tid] = b1[tid];
  if (tid < Hh) { sB2[tid] = b2[tid]; sB3[tid] = b3[tid]; sWu0[tid] = Wu[tid]; }

  // y[n][k] = trace[b,n,:] . W1[k,:]   (layer-1 factorization)
  for (int i = tid; i < Nn * H1; i += 256) {
    int n = i >> 5, k = i & 31;
    const float* tr = &trace[(b * Nn + n) * Tt];
    sY[i] = tr[0] * W1[k * Tt + 0] + tr[1] * W1[k * Tt + 1];
  }
  // hu1[n] = h[n].wu_B ; hu2[n] = h[n].wu_C
  {
    int n = tid & 127;
    const float* hv = &hid[(b * Nn + n) * Hh];
    const float* wv = (tid < 128) ? (Wu + 64) : (Wu + 128);
    float s = 0.f;
    for (int k = 0; k < Hh; ++k) s += hv[k] * wv[k];
    if (tid < 128) sHu1[n] = s; else sHu2[n] = s;
  }
  // gbP[pp][j] = bg[j] + h[p0+pp] . Gb[j,:]
  for (int i = tid; i < 16 * Hh; i += 256) {
    int pp = i >> 6, j = i & 63;
    const float* hv = &hid[(b * Nn + p0 + pp) * Hh];
    float s = bg[j];
    for (int k = 0; k < Hh; ++k) s += hv[k] * Wg[j * 192 + 64 + k];
    sGbP[i] = s;
  }
  for (int i = tid; i < 16 * Hh; i += 256) sAgg[i] = 0.f;
  __syncthreads();

  // hloc for one (pp, l-tile): layer2 (K=32) + layer3 (K=64), WMMA-chained.
  auto compute_hloc = [&](int pn, int l0, v8f* acc3) {
    // layer2 A operand: relu(y[p]-y[l]+b1), built directly in A-layout
    v2f a2[8];
#pragma unroll
    for (int c = 0; c < 8; ++c) {
      int k0 = 4 * c + kb;
      float x0 = sY[pn * H1 + k0]     - sY[(l0 + m) * H1 + k0]     + sB1[k0];
      float x1 = sY[pn * H1 + k0 + 1] - sY[(l0 + m) * H1 + k0 + 1] + sB1[k0 + 1];
      a2[c].x = fmaxf(x0, 0.f);
      a2[c].y = fmaxf(x1, 0.f);
    }
#pragma unroll
    for (int nc = 0; nc < 4; ++nc) {
      int col = nc * 16 + m;
      float bb = sB2[col];
      v8f acc = {bb, bb, bb, bb, bb, bb, bb, bb};
#pragma unroll
      for (int c = 0; c < 8; ++c) {
        v2f bv;
        bv.x = sW2[col * H1 + 4 * c + kb];
        bv.y = sW2[col * H1 + 4 * c + kb + 1];
        acc = WMMA_F32(a2[c], bv, acc);
      }
#pragma unroll
      for (int g = 0; g < 8; ++g)
        stage[(g + 8 * hi) * Hh + col] = fmaxf(acc[g], 0.f);  // relu -> x2
    }
    asm volatile("s_wait_dscnt 0x0" ::: "memory");
    // layer3: hloc = x2 @ W3^T + b3  (K=64)
    v2f a3[16];
#pragma unroll
    for (int c = 0; c < 16; ++c) {
      a3[c].x = stage[m * Hh + 4 * c + kb];
      a3[c].y = stage[m * Hh + 4 * c + kb + 1];
    }
#pragma unroll
    for (int nc = 0; nc < 4; ++nc) {
      int col = nc * 16 + m;
      float bb = sB3[col];
      v8f acc = {bb, bb, bb, bb, bb, bb, bb, bb};
#pragma unroll
      for (int c = 0; c < 16; ++c) {
        v2f bv;
        bv.x = sW3[col * Hh + 4 * c + kb];
        bv.y = sW3[col * Hh + 4 * c + kb + 1];
        acc = WMMA_F32(a3[c], bv, acc);
      }
      acc3[nc] = acc;
    }
#pragma unroll
    for (int nc = 0; nc < 4; ++nc) {
      int col = nc * 16 + m;
#pragma unroll
      for (int g = 0; g < 8; ++g)
        stage[(g + 8 * hi) * Hh + col] = acc3[nc][g];  // stage hloc
    }
    asm volatile("s_wait_dscnt 0x0" ::: "memory");
  };

  // ---------------- pass A: u[p][l] for all l ----------------
  for (int lt = 0; lt < 8; ++lt) {
    int l0 = lt * 16;
    for (int sub = 0; sub < 2; ++sub) {
      int pp = wave * 2 + sub;
      int pn = p0 + pp;
      v8f acc3[4];
      compute_hloc(pn, l0, acc3);
      if (lane < 16) {
        float u = sHu1[pn] + sHu2[l0 + lane];
#pragma unroll 8
        for (int k = 0; k < Hh; ++k) u += stage[lane * Hh + k] * sWu0[k];
        sU[pp * Nn + l0 + lane] = u;
      }
    }
  }
  __syncthreads();

  // softmax over l (128-wide), one row per thread
  if (tid < 16) {
    float mx = -1e30f;
    for (int l = 0; l < Nn; ++l) mx = fmaxf(mx, sU[tid * Nn + l]);
    float s = 0.f;
    for (int l = 0; l < Nn; ++l) {
      float e = expf(sU[tid * Nn + l] - mx);
      sU[tid * Nn + l] = e;
      s += e;
    }
    float inv = 1.f / s;
    for (int l = 0; l < Nn; ++l) sU[tid * Nn + l] *= inv;
  }
  __syncthreads();

  // ---------------- pass B: gate + alpha-weighted aggregation --------------
  for (int lt = 0; lt < 8; ++lt) {
    int l0 = lt * 16;
    __syncthreads();  // protect sHl/sGcL from prior iteration readers
    for (int i = tid; i < 16 * Hh; i += 256)
      sHl[i] = hid[(b * Nn + l0 + (i >> 6)) * Hh + (i & 63)];
    __syncthreads();
    for (int i = tid; i < 16 * Hh; i += 256) {
      int ll = i >> 6, j = i & 63;
      float s = 0.f;
      for (int k = 0; k < Hh; ++k) s += sHl[ll * Hh + k] * Wg[j * 192 + 128 + k];
      sGcL[i] = s;
    }
    __syncthreads();

    for (int sub = 0; sub < 2; ++sub) {
      int pp = wave * 2 + sub;
      int pn = p0 + pp;
      v8f acc3[4];
      compute_hloc(pn, l0, acc3);
      // gate GEMM: hloc @ Ga^T  (K=64), C preloaded with gb[p]+gc[l]
      v2f aG[16];
#pragma unroll
      for (int c = 0; c < 16; ++c) {
        aG[c].x = stage[m * Hh + 4 * c + kb];
        aG[c].y = stage[m * Hh + 4 * c + kb + 1];
      }
#pragma unroll
      for (int nc = 0; nc < 4; ++nc) {
        int col = nc * 16 + m;
        v8f acc;
#pragma unroll
        for (int g = 0; g < 8; ++g)
          acc[g] = sGbP[pp * Hh + col] + sGcL[(g + 8 * hi) * Hh + col];
#pragma unroll
        for (int c = 0; c < 16; ++c) {
          v2f bv;
          bv.x = sGa[col * Hh + 4 * c + kb];
          bv.y = sGa[col * Hh + 4 * c + kb + 1];
          acc = WMMA_F32(aG[c], bv, acc);
        }
        float part = 0.f;
#pragma unroll
        for (int g = 0; g < 8; ++g) {
          int row = g + 8 * hi;
          float gate = 1.f / (1.f + expf(-acc[g]));
          part += sU[pp * Nn + l0 + row] * gate * sHl[row * Hh + col];
        }
        atomicAdd(&sAgg[pp * Hh + col], part);  // ds_add_f32
      }
    }
  }
  __syncthreads();

  // ---------------- msg = agg @ Wc^T ; out = cell + 2*msg ------------------
  for (int i = tid; i < 16 * Hh; i += 256) {
    int pp = i >> 6, j = i & 63;
    float s = 0.f;
    for (int k = 0; k < Hh; ++k) s += sAgg[pp * Hh + k] * Wc[j * Hh + k];
    int idx = (b * Nn + p0 + pp) * Hh + j;
    out[idx] = cell[idx] + 2.f * s;  // L=2 residual adds
  }
}

extern "C" void kernel_launch(void* const* d_in, const int* in_sizes, int n_in,
                              void* d_out, int out_size, void* d_ws, size_t ws_size,
                              hipStream_t stream) {
  (void)in_sizes; (void)n_in; (void)d_ws; (void)ws_size; (void)out_size;
  // inputs: traces, cell, hidden, W1,b1, W2,b2, W3,b3, Wu, Wg, bg, Wc
  srmg_kernel<<<dim3(32 * 8), dim3(256), 0, stream>>>(
      (const float*)d_in[0], (const float*)d_in[1], (const float*)d_in[2],
      (const float*)d_in[3], (const float*)d_in[4], (const float*)d_in[5],
      (const float*)d_in[6], (const float*)d_in[7], (const float*)d_in[8],
      (const float*)d_in[9], (const float*)d_in[10], (const float*)d_in[11],
      (const float*)d_in[12], (float*)d_out);
}